// ObservationEmbedding_7825430413654
// MI455X (gfx1250) — compile-verified
//
#include <hip/hip_runtime.h>

// MI455X / gfx1250, wave32. Fused single-pass kernel.
// out[B*T, 256] f32 = [ features@W^T + b  (128 cols, via V_WMMA_F32_16X16X4_F32) |
//                       sinusoidal PE      (128 cols, 64x128 LDS table)          ]
// Roofline: 64 MB read + 1 GB write => ~46us floor at 23.3 TB/s; store-bound.
// Therefore: every global store is a b128 (512B contiguous per wave-instruction),
// GEMM results are transposed from WMMA D-layout to row-major via an LDS stage
// (row stride 132 floats => halves of the wave hit disjoint bank ranges).

typedef __attribute__((ext_vector_type(2))) float v2f;
typedef __attribute__((ext_vector_type(4))) float v4f;
typedef __attribute__((ext_vector_type(8))) float v8f;

#define B_DIM   16384
#define T_DIM   64
#define F_DIM   16
#define DH      128    // D/2
#define D_OUT   256
#define M_ROWS  (B_DIM * T_DIM)                  // 1,048,576
#define M_TILES (M_ROWS / 16)                    // 65,536
#define WAVES_PER_BLOCK 8
#define GRID_BLOCKS (M_TILES / WAVES_PER_BLOCK)  // 8,192
#define ROW_PAD 132   // 16x128 stage with padded stride: bank-conflict-free, 16B-aligned rows

__global__ __launch_bounds__(256)
void obs_embed_fused_kernel(const float* __restrict__ feat,   // [M_ROWS,16]
                            const float* __restrict__ W,      // [128,16] torch layout
                            const float* __restrict__ bias,   // [128]
                            const int*   __restrict__ pos,    // [64]
                            float* __restrict__ out)          // [M_ROWS,256]
{
    __shared__ __align__(16) float sW[DH * F_DIM];                     //  8 KB
    __shared__ float sB[DH];                                           // .5 KB
    __shared__ __align__(16) float sPE[T_DIM * DH];                    // 32 KB
    __shared__ __align__(16) float sStage[WAVES_PER_BLOCK][16 * ROW_PAD]; // 66 KB

    // ---- Phase 0: stage W/bias, build the 64x128 PE table (once per block) ----
    for (int i = threadIdx.x; i < DH * F_DIM; i += 256) sW[i] = W[i];
    if (threadIdx.x < DH) sB[threadIdx.x] = bias[threadIdx.x];

    const float LN1E4_OVER_128 = 0.07195578492459019f;  // ln(10000)/128
    for (int idx = threadIdx.x; idx < T_DIM * DH; idx += 256) {
        const int t = idx >> 7;
        const int d = idx & (DH - 1);
        const int j = d >> 1;
        const float inv_freq = __expf(-(float)(4 * j) * LN1E4_OVER_128);
        const float ang = (float)pos[t] * inv_freq;
        sPE[idx] = (d & 1) ? cosf(ang) : sinf(ang);
    }
    __syncthreads();

    const int wave    = threadIdx.x >> 5;
    const int lane    = threadIdx.x & 31;
    const int halfSel = lane >> 4;          // 0: lanes 0-15, 1: lanes 16-31
    const int lrow    = lane & 15;

    const long tileM = (long)blockIdx.x * WAVES_PER_BLOCK + wave;   // < 65536
    const long row0  = tileM * 16;
    const int  t0    = (int)(row0 & (T_DIM - 1));   // in {0,16,32,48}; t0+15 < 64
    float* stage = &sStage[wave][0];

    // ---- Phase 1: WMMA GEMM, 16 rows x 128 cols per wave ----
    // A fragment layout (ISA 32-bit A 16x4): lane holds K = 4g + 2*halfSel + {0,1}
    v2f a[4];
    const float* arow = feat + (row0 + lrow) * F_DIM;
#pragma unroll
    for (int g = 0; g < 4; ++g)
        a[g] = *(const v2f*)(arow + 4 * g + 2 * halfSel);   // 8B coalesced

#pragma unroll
    for (int n = 0; n < 8; ++n) {
        const int col = 16 * n + lrow;      // B/C/D: lane -> column
        v8f c = {};
#pragma unroll
        for (int g = 0; g < 4; ++g) {
            // B = W^T: B[k][col] = W[col][k]; same contiguous-pair layout as A.
            v2f bf = *(const v2f*)(&sW[col * F_DIM + 4 * g + 2 * halfSel]);
            c = __builtin_amdgcn_wmma_f32_16x16x4_f32(
                    false, a[g], false, bf, (short)0, c, false, false);
        }
        const float bv = sB[col];
        // D layout -> LDS stage: VGPR j is row (j + 8*halfSel), lane&15 is column.
        // ROW_PAD=132: rows j and j+8 differ by 32 banks => no half-wave conflict.
#pragma unroll
        for (int j = 0; j < 8; ++j)
            stage[(8 * halfSel + j) * ROW_PAD + col] = c[j] + bv;
    }

    // ---- Phase 2: stream 16 full output rows with b128 stores ----
    // Per row: 128 GEMM floats = 32 float4 = one ds_load_b128 + global_store_b128
    // across the wave (512B contiguous), same for the PE half. 1KB row in 2 stores.
#pragma unroll
    for (int r = 0; r < 16; ++r) {
        float* orow = out + (row0 + r) * D_OUT;
        v4f g = *(const v4f*)(stage + r * ROW_PAD + 4 * lane);
        *(v4f*)(orow + 4 * lane) = g;
        v4f p = *(const v4f*)(sPE + (t0 + r) * DH + 4 * lane);
        *(v4f*)(orow + DH + 4 * lane) = p;
    }
}

// ---------------------------------------------------------------------------
extern "C" void kernel_launch(void* const* d_in, const int* in_sizes, int n_in,
                              void* d_out, int out_size, void* d_ws, size_t ws_size,
                              hipStream_t stream) {
    const float* feat = (const float*)d_in[0];   // [16384,64,16]
    const float* W    = (const float*)d_in[1];   // [128,16]
    const float* bias = (const float*)d_in[2];   // [128]
    const int*   pos  = (const int*)d_in[3];     // [64]
    float* out = (float*)d_out;                  // [16384,64,256]

    obs_embed_fused_kernel<<<GRID_BLOCKS, 256, 0, stream>>>(feat, W, bias, pos, out);
}